// StressPDE_54941221650894
// MI455X (gfx1250) — compile-verified
//
#include <hip/hip_runtime.h>

// ---------------------------------------------------------------------------
// StressPDE for MI455X (gfx1250)
//   out[i] = reaction[i] + 0.1 * sum_j adj[i][j] * exp(-(S[j]-S[i])^2/sigma^2) * (S[j]-S[i])
//   reaction = tanh([S,t] @ W1 + b1) @ W2 + b2
//
// Roofline: adj = 256 MB fp32 streamed once -> ~11us @ 23.3 TB/s; compute is
// ~0.4 GFLOP -> bandwidth bound. Kernel 2 streams adj with coalesced,
// NON-TEMPORAL b128 loads (one-pass stream must not displace L2). S (32 KB)
// is DMA'd into LDS per block via the Tensor Data Mover. Kernel 1 does the
// tiny MLP with V_WMMA_F32_16X16X4_F32 (K=2 padded to 4) and native V_TANH.
// Diagonal zeroing is free: S_diff==0 at j==i makes the term exactly 0.
// ---------------------------------------------------------------------------

typedef __attribute__((ext_vector_type(2))) float    v2f;
typedef __attribute__((ext_vector_type(4))) float    v4f;
typedef __attribute__((ext_vector_type(8))) float    v8f;
typedef __attribute__((ext_vector_type(4))) unsigned u32x4;
typedef __attribute__((ext_vector_type(4))) int      i32x4;
typedef __attribute__((ext_vector_type(8))) int      i32x8;

#define NU      8192
#define HID     64
#define SIGMA2  2.25f           // 1.5^2
#define NEGINV  (-1.0f / SIGMA2)
#define DCOEFF  0.1f

// Native CDNA5 V_TANH_F32 if the toolchain exposes it; else libm fallback.
#if __has_builtin(__builtin_amdgcn_tanhf)
  #define FAST_TANH(x) __builtin_amdgcn_tanhf(x)
#elif __has_builtin(__builtin_amdgcn_tanh_f32)
  #define FAST_TANH(x) __builtin_amdgcn_tanh_f32(x)
#else
  #define FAST_TANH(x) tanhf(x)
#endif

// ---------------------------------------------------------------------------
// Kernel 1: reaction MLP via WMMA f32 16x16x4.
// One wave per 16 rows. A = [S_m, t, 0, 0] (16x4), B = W1 slice (4x16),
// C = b1 broadcast. 4 WMMA tiles cover HID=64 columns, then tanh * W2 and a
// 16-lane shfl_xor reduction produces the per-row scalar.
// ---------------------------------------------------------------------------
__global__ __launch_bounds__(256)
void reaction_wmma_kernel(const float* __restrict__ t,
                          const float* __restrict__ S,
                          const float* __restrict__ W1,   // (2, 64) row-major
                          const float* __restrict__ b1,   // (64)
                          const float* __restrict__ W2,   // (64, 1)
                          const float* __restrict__ b2,   // (1)
                          float* __restrict__ reaction,
                          int n) {
  const int wave = (blockIdx.x * blockDim.x + threadIdx.x) >> 5;
  const int lane = threadIdx.x & 31;
  const int i0 = wave * 16;
  if (i0 >= n) return;                 // wave-uniform: EXEC stays all-1s

  const bool  lo   = (lane < 16);
  const int   nsel = lane & 15;        // column (hidden index mod 16) this lane owns
  const float tval = t[0];

  // A-matrix 16x4 f32 layout: lanes 0-15 hold K=0 (v0) and K=1 (v1);
  // lanes 16-31 hold K=2,3 which we zero-pad (feat has K=2).
  v2f a;
  a.x = lo ? S[i0 + lane] : 0.0f;
  a.y = lo ? tval : 0.0f;

  float acc[8];
#pragma unroll
  for (int r = 0; r < 8; ++r) acc[r] = 0.0f;

#pragma unroll
  for (int tile = 0; tile < 4; ++tile) {
    const int h0 = tile * 16;
    // B-matrix 4x16: row K striped across lanes; K=0 -> W1 row 0, K=1 -> W1 row 1,
    // K=2,3 (lanes 16-31) zero.
    v2f b;
    b.x = lo ? W1[h0 + nsel]       : 0.0f;
    b.y = lo ? W1[HID + h0 + nsel] : 0.0f;
    const float cval = b1[h0 + nsel];            // C[m][n] = b1[h0+n], all m
    v8f c = {cval, cval, cval, cval, cval, cval, cval, cval};
    // D = A*B + C  -> v_wmma_f32_16x16x4_f32
    v8f d = __builtin_amdgcn_wmma_f32_16x16x4_f32(
        /*neg_a=*/false, a, /*neg_b=*/false, b,
        /*c_mod=*/(short)0, c, /*reuse_a=*/false, /*reuse_b=*/false);
    const float w2v = W2[h0 + nsel];
#pragma unroll
    for (int r = 0; r < 8; ++r)
      acc[r] += FAST_TANH(d[r]) * w2v;           // hidden h = h0 + nsel, row M = r(+8)
  }

  // Sum over the 16 hidden-columns owned by each 16-lane half.
#pragma unroll
  for (int m = 8; m >= 1; m >>= 1) {
#pragma unroll
    for (int r = 0; r < 8; ++r) acc[r] += __shfl_xor(acc[r], m, 32);
  }

  const float bb = b2[0];
  if (lane == 0) {
#pragma unroll
    for (int r = 0; r < 8; ++r) reaction[i0 + r] = acc[r] + bb;       // M = r
  }
  if (lane == 16) {
#pragma unroll
    for (int r = 0; r < 8; ++r) reaction[i0 + 8 + r] = acc[r] + bb;   // M = 8 + r
  }
}

// ---------------------------------------------------------------------------
// Kernel 2: bandwidth-bound diffusion stream.
// 8 waves/block, one row per wave, 1024 blocks. S (32 KB) is loaded into LDS
// once per block by the Tensor Data Mover (1-D descriptor), waited on with
// s_wait_tensorcnt, then all waves stream their adj row with non-temporal
// float4 loads (one-pass 256 MB stream; keep it out of L2).
// ---------------------------------------------------------------------------
__global__ __launch_bounds__(256)
void diffusion_kernel(const float* __restrict__ S,
                      const float* __restrict__ adj,
                      const float* __restrict__ reaction,
                      float* __restrict__ out,
                      int n) {
  __shared__ float sS[NU];

  if (threadIdx.x < 32) {   // wave 0 issues the TDM DMA (EXEC-independent op)
    const unsigned lds_off = (unsigned)(size_t)(&sS[0]);
    const unsigned long long ga = (unsigned long long)(size_t)S;
    // D# group 0: count=1 | lds_addr | global_addr[56:0] | type=2
    u32x4 g0;
    g0[0] = 1u;                                   // count = 1 valid descriptor
    g0[1] = lds_off;                              // lds_addr
    g0[2] = (unsigned)(ga & 0xFFFFFFFFu);         // global_addr[31:0]
    g0[3] = (unsigned)((ga >> 32) & 0x1FFFFFFu)   // global_addr[56:32]
          | (2u << 30);                           // type = 2 ("image")
    // D# group 1: data_size=4B, tensor_dim0 = tile_dim0 = n (1-D), stride0 = n
    i32x8 g1;
    g1[0] = (int)(2u << 16);                      // data_size = 2 -> 4 bytes
    g1[1] = (int)(((unsigned)n & 0xFFFFu) << 16); // tensor_dim0[15:0] @ bits63:48
    g1[2] = (int)((((unsigned)n >> 16) & 0xFFFFu) // tensor_dim0[31:16]
          | (1u << 16));                          // tensor_dim1 = 1
    g1[3] = (int)(((unsigned)n & 0xFFFFu) << 16); // tile_dim0 = n @ bits127:112
    g1[4] = 0;                                    // tile_dim1/2 unused
    g1[5] = n;                                    // tensor_dim0_stride[31:0]
    g1[6] = 0;
    g1[7] = 0;
    i32x4 gz4 = {0, 0, 0, 0};
#if defined(__clang_major__) && __clang_major__ >= 23
    i32x8 gz8 = {0, 0, 0, 0, 0, 0, 0, 0};
    __builtin_amdgcn_tensor_load_to_lds(g0, g1, gz4, gz4, gz8, 0);
#else
    __builtin_amdgcn_tensor_load_to_lds(g0, g1, gz4, gz4, 0);
#endif
    __builtin_amdgcn_s_wait_tensorcnt(0);
  }
  __syncthreads();

  const int wave = threadIdx.x >> 5;
  const int lane = threadIdx.x & 31;
  const int i = blockIdx.x * 8 + wave;
  if (i >= n) return;

  const float si = sS[i];
  const v4f* __restrict__ row4 = (const v4f*)(adj + (size_t)i * (size_t)n);
  const v4f* __restrict__ s4 = (const v4f*)sS;
  const int nq = n >> 2;                          // 2048 float4 per row

  float acc = 0.0f;
#pragma unroll 4
  for (int q = lane; q < nq; q += 32) {
    if (q + 128 < nq) __builtin_prefetch(row4 + q + 128, 0, 0); // global_prefetch_b8
    const v4f a4 = __builtin_nontemporal_load(row4 + q); // NT b128 stream of adj
    const v4f sv = s4[q];                                // ds_load_b128 of S[j..j+3]
    float d0 = sv.x - si;  acc += a4.x * __expf(d0 * d0 * NEGINV) * d0;
    float d1 = sv.y - si;  acc += a4.y * __expf(d1 * d1 * NEGINV) * d1;
    float d2 = sv.z - si;  acc += a4.z * __expf(d2 * d2 * NEGINV) * d2;
    float d3 = sv.w - si;  acc += a4.w * __expf(d3 * d3 * NEGINV) * d3;
  }
  acc *= DCOEFF;

  // wave32 cross-lane row-sum
#pragma unroll
  for (int m = 16; m >= 1; m >>= 1) acc += __shfl_xor(acc, m, 32);

  if (lane == 0) out[i] = acc + reaction[i];
}

// ---------------------------------------------------------------------------
// Host launcher. Inputs: t, S_group, adj, W1, b1, W2, b2. Output: (n,1) f32.
// d_ws holds the intermediate reaction vector (n floats = 32 KB).
// ---------------------------------------------------------------------------
extern "C" void kernel_launch(void* const* d_in, const int* in_sizes, int n_in,
                              void* d_out, int out_size, void* d_ws, size_t ws_size,
                              hipStream_t stream) {
  const float* t   = (const float*)d_in[0];
  const float* S   = (const float*)d_in[1];
  const float* adj = (const float*)d_in[2];
  const float* W1  = (const float*)d_in[3];
  const float* b1  = (const float*)d_in[4];
  const float* W2  = (const float*)d_in[5];
  const float* b2  = (const float*)d_in[6];
  const int n = in_sizes[1];                     // 8192
  float* reaction = (float*)d_ws;
  float* out = (float*)d_out;

  // Kernel 1: one wave per 16 rows -> n/16 waves, 8 waves/block.
  const int rblocks = (n / 16 + 7) / 8;
  reaction_wmma_kernel<<<rblocks, 256, 0, stream>>>(t, S, W1, b1, W2, b2,
                                                    reaction, n);
  // Kernel 2: one wave per row, 8 rows/block.
  diffusion_kernel<<<(n + 7) / 8, 256, 0, stream>>>(S, adj, reaction, out, n);
}